// PyramidROIAlign_AFP_32091995636089
// MI455X (gfx1250) — compile-verified
//
#include <hip/hip_runtime.h>

// PyramidROIAlign (crop_and_resize 7x7, bilinear, extrapolation=0) over 4 FPN
// levels. Memory-bound: feature maps (178 MB) fit MI455X's 192 MB L2, so the
// gather reads hit L2; output stores (205 MB) are streaming and marked
// non-temporal (th:TH_STORE_NT) so they don't evict the read working set.
// HBM traffic ~383 MB -> ~16us floor at 23.3 TB/s.
// CDNA5 path: GLOBAL_LOAD_ASYNC_TO_LDS_B128 (ASYNCcnt DMA engine) with a
// double-buffered LDS pipeline, s_wait_asynccnt / s_wait_dscnt fencing.
// Block = one (b, n, py) output row: box + y-side math hoisted out of the
// px loop; remaining per-px math is block-uniform -> scalar unit.

static constexpr int Bn = 2;
static constexpr int Nn = 512;
static constexpr int Cn = 256;   // channels (contiguous, NHWC)
static constexpr int PH = 7;
static constexpr int PW = 7;

typedef float v4f __attribute__((ext_vector_type(4)));

// ---- CDNA5 async global->LDS copy (16B per lane), inline asm ---------------
__device__ __forceinline__ void async_copy_b128(unsigned lds_off, const float* gsrc) {
    asm volatile("global_load_async_to_lds_b128 %0, %1, off"
                 :: "v"(lds_off), "v"((unsigned long long)(const void*)gsrc)
                 : "memory");
}
__device__ __forceinline__ void wait_async_le4() {
    asm volatile("s_wait_asynccnt 0x4" ::: "memory");
}
__device__ __forceinline__ void wait_async0() {
    asm volatile("s_wait_asynccnt 0x0" ::: "memory");
}
__device__ __forceinline__ void wait_ds0() {
    asm volatile("s_wait_dscnt 0x0" ::: "memory");
}

__global__ __launch_bounds__(64)
void roialign_row_kernel(const float* __restrict__ boxes,
                         const float* __restrict__ feat,
                         float* __restrict__ out, int H) {
    // 2 buffers x 4 corner rows x 256 floats = 8 KB LDS
    __shared__ __align__(16) float smem[2 * 4 * Cn];

    const int t = threadIdx.x;                        // 0..63, owns channels [4t,4t+4)
    const unsigned ldsBase = (unsigned)(unsigned long long)(void*)smem;

    // ---- per-block (row) decode: one (b, n, py) row of 7 positions ----------
    const int row = blockIdx.x;                       // 0..7167
    const int py  = row % PH;
    const int bn  = row / PH;                         // 0..1023 == b*Nn + n
    const int b   = bn / Nn;

    const float hm1 = (float)(H - 1);
    const float* bx = boxes + (size_t)bn * 4;
    const float y1 = bx[0], x1 = bx[1], y2 = bx[2], x2 = bx[3];

    // y-side: invariant across the row
    const float ty  = (float)py * (1.0f / 6.0f);
    const float y   = (y1 + ty * (y2 - y1)) * hm1;
    const bool  vy  = (y >= 0.0f) && (y <= hm1);
    const float y0f = floorf(y);
    const float ly  = y - y0f;
    const int   y0  = min(max((int)y0f, 0), H - 1);
    const int   yb  = min(y0 + 1, H - 1);

    const float* img     = feat + (size_t)b * H * H * Cn;
    const float* topBase = img + (size_t)y0 * H * Cn;   // row y0, column 0
    const float* botBase = img + (size_t)yb * H * Cn;   // row yb, column 0
    float* orow = out + ((size_t)bn * (PH * PW) + (size_t)py * PW) * Cn + t * 4;

    const float dx  = (x2 - x1) * hm1;
    const float x1s = x1 * hm1;

    // x-side state for one px (uniform across the block -> scalar regs)
    float lx_c, lx_n, vld_c, vld_n;
    int   xo_c[2], xo_n[2];          // x0*Cn, xb*Cn element offsets

    auto setup_x = [&](int px, float& lx, float& vld, int* xo) {
        const float x   = x1s + (float)px * (1.0f / 6.0f) * dx;
        const bool  vx  = (x >= 0.0f) && (x <= hm1);
        const float x0f = floorf(x);
        lx  = x - x0f;
        vld = (vy && vx) ? 1.0f : 0.0f;
        const int x0 = min(max((int)x0f, 0), H - 1);
        const int xb = min(x0 + 1, H - 1);
        xo[0] = x0 * Cn;
        xo[1] = xb * Cn;
    };

    // issue 4 x 16B async copies (this lane's channel slice of 4 corner rows)
    auto prefetch = [&](const int* xo, int buf) {
        const unsigned o = ldsBase + (unsigned)(buf * 4 * Cn * 4) + (unsigned)(t * 16);
        const int c = t * 4;
        async_copy_b128(o + 0u * (Cn * 4), topBase + xo[0] + c);
        async_copy_b128(o + 1u * (Cn * 4), topBase + xo[1] + c);
        async_copy_b128(o + 2u * (Cn * 4), botBase + xo[0] + c);
        async_copy_b128(o + 3u * (Cn * 4), botBase + xo[1] + c);
    };

    setup_x(0, lx_c, vld_c, xo_c);
    prefetch(xo_c, 0);

    #pragma unroll
    for (int px = 0; px < PW; ++px) {
        const int buf = px & 1;
        if (px + 1 < PW) {
            setup_x(px + 1, lx_n, vld_n, xo_n);
            wait_ds0();                 // drain prior LDS reads of buf^1 (WAR)
            prefetch(xo_n, buf ^ 1);
            wait_async_le4();           // in-order: buf's 4 copies now resident
        } else {
            wait_async0();
        }

        const float* s = smem + buf * 4 * Cn + t * 4;
        v4f v00 = *(const v4f*)(s + 0 * Cn);
        v4f v01 = *(const v4f*)(s + 1 * Cn);
        v4f v10 = *(const v4f*)(s + 2 * Cn);
        v4f v11 = *(const v4f*)(s + 3 * Cn);

        const float lx = lx_c, vld = vld_c;
        auto blend = [&](float a, float bb, float c, float d) {
            float top = a + (bb - a) * lx;
            float bot = c + (d - c) * lx;
            return (top + (bot - top) * ly) * vld;
        };
        v4f o;
        o.x = blend(v00.x, v01.x, v10.x, v11.x);
        o.y = blend(v00.y, v01.y, v10.y, v11.y);
        o.z = blend(v00.z, v01.z, v10.z, v11.z);
        o.w = blend(v00.w, v01.w, v10.w, v11.w);

        // streaming output: non-temporal 128-bit store, keep L2 for gathers
        __builtin_nontemporal_store(o, (v4f*)(orow + (size_t)px * Cn));

        lx_c = lx_n; vld_c = vld_n; xo_c[0] = xo_n[0]; xo_c[1] = xo_n[1];
    }
}

extern "C" void kernel_launch(void* const* d_in, const int* in_sizes, int n_in,
                              void* d_out, int out_size, void* d_ws, size_t ws_size,
                              hipStream_t stream) {
    (void)in_sizes; (void)n_in; (void)out_size; (void)d_ws; (void)ws_size;

    const float* boxes = (const float*)d_in[0];
    const float* f2    = (const float*)d_in[1];   // [2,256,256,256]
    const float* f3    = (const float*)d_in[2];   // [2,128,128,256]
    const float* f4    = (const float*)d_in[3];   // [2, 64, 64,256]
    const float* f5    = (const float*)d_in[4];   // [2, 32, 32,256]
    float* out = (float*)d_out;

    const size_t lvl_stride = (size_t)Bn * Nn * PH * PW * Cn;  // 12,845,056
    const int rows = Bn * Nn * PH;                             // 7,168 blocks
    dim3 grid(rows);
    dim3 block(64);

    roialign_row_kernel<<<grid, block, 0, stream>>>(boxes, f2, out + 0 * lvl_stride, 256);
    roialign_row_kernel<<<grid, block, 0, stream>>>(boxes, f3, out + 1 * lvl_stride, 128);
    roialign_row_kernel<<<grid, block, 0, stream>>>(boxes, f4, out + 2 * lvl_stride, 64);
    roialign_row_kernel<<<grid, block, 0, stream>>>(boxes, f5, out + 3 * lvl_stride, 32);
}